// DipolePredictorGCN_66331474919538
// MI455X (gfx1250) — compile-verified
//
#include <hip/hip_runtime.h>
#include <hip/hip_bf16.h>

typedef float v2f __attribute__((ext_vector_type(2)));
typedef float v8f __attribute__((ext_vector_type(8)));

// ---------------------------------------------------------------------------
// Workspace layout (floats):
//   deg   [N]      node degree (edge count into dst)
//   s1    [N]      scalar layer-1 aggregate: sum over edges of x[src]*norm
//   dis   [N]      deg_inv_sqrt
//   t     [N]      s1 + dis^2 * x   (scalar pre-activation seed for layer 1)
//   h2pre [N*32]   relu(t*W1+b1) @ W2   (WMMA output)
//   agg2  [N*32]   layer-2 edge aggregate
//   psum  [G*32]   pooled sums
//   pcnt  [G]      pooled counts
// ---------------------------------------------------------------------------

__global__ void gcn_zero_f32(float* __restrict__ p, long n) {
    long i = (long)blockIdx.x * blockDim.x + threadIdx.x;
    long stride = (long)gridDim.x * blockDim.x;
    for (; i < n; i += stride) p[i] = 0.0f;
}

__global__ void gcn_degree(const int* __restrict__ dst, float* __restrict__ deg, int E) {
    int e = blockIdx.x * blockDim.x + threadIdx.x;
    if (e < E) atomicAdd(&deg[dst[e]], 1.0f);
}

__global__ void gcn_dis(const float* __restrict__ deg, float* __restrict__ dis, int N) {
    int i = blockIdx.x * blockDim.x + threadIdx.x;
    if (i < N) dis[i] = rsqrtf(deg[i] + 1.0f);
}

// Layer-1 edge aggregation collapsed to a scalar per edge (x is [N,1]):
// s1[dst] += x[src] * dis[src] * dis[dst]
__global__ void gcn_edge1(const int* __restrict__ src, const int* __restrict__ dst,
                          const float* __restrict__ x, const float* __restrict__ dis,
                          float* __restrict__ s1, int E) {
    int e = blockIdx.x * blockDim.x + threadIdx.x;
    if (e < E) {
        int s = src[e], d = dst[e];
        atomicAdd(&s1[d], x[s] * dis[s] * dis[d]);
    }
}

__global__ void gcn_make_t(const float* __restrict__ s1, const float* __restrict__ dis,
                           const float* __restrict__ x, float* __restrict__ t, int N) {
    int i = blockIdx.x * blockDim.x + threadIdx.x;
    if (i < N) t[i] = s1[i] + dis[i] * dis[i] * x[i];
}

// ---------------------------------------------------------------------------
// Fused layer-1 epilogue + layer-2 dense GEMM via V_WMMA_F32_16X16X4_F32.
//   h1[m,k] = relu(t[m]*W1[k] + b1[k])   (generated on the fly, never stored)
//   h2pre   = h1 @ W2                    (M=100000, K=64, N=32)
// One wave computes one 16x16 output tile; K marched in 16 steps of 4.
// A-frag (f32 16x4): lanes 0-15 hold M=lane, K={k0,k0+1}; lanes 16-31 K={k0+2,k0+3}.
// B-frag (f32 4x16): lanes 0-15 hold N=lane, rows {k0,k0+1}; lanes 16-31 rows {k0+2,k0+3}.
// D (16x16 f32, 8 VGPRs): vgpr r -> row m0+r (lanes 0-15) / m0+r+8 (lanes 16-31).
// ---------------------------------------------------------------------------
__global__ void gcn_gemm_wmma(const float* __restrict__ t,
                              const float* __restrict__ W1,   // [64]
                              const float* __restrict__ b1,   // [64]
                              const float* __restrict__ W2,   // [64 x 32] row-major
                              float* __restrict__ h2pre,      // [N x 32]
                              int nTiles) {
    int wave = (int)((blockIdx.x * blockDim.x + threadIdx.x) >> 5);
    int lane = threadIdx.x & 31;
    if (wave >= nTiles) return;          // wave-uniform: EXEC stays all-ones

    int mt = wave >> 1;                  // 16-row tile index
    int nt = wave & 1;                   // 16-col tile index (N=32 -> 2 tiles)
    int m0 = mt << 4;
    int n0 = nt << 4;
    int r = lane & 15;
    int khalf = lane >> 4;               // 0: K pair {0,1}; 1: K pair {2,3}

    float trow = t[m0 + r];              // shared by A-frag rows

    v8f acc = {};
#pragma unroll
    for (int k0 = 0; k0 < 64; k0 += 4) {
        int ka = k0 + khalf * 2;
        // on-the-fly A: relu(t*W1 + b1)
        float a0 = fmaf(trow, W1[ka],     b1[ka]);
        float a1 = fmaf(trow, W1[ka + 1], b1[ka + 1]);
        v2f a, b;
        a.x = a0 > 0.0f ? a0 : 0.0f;
        a.y = a1 > 0.0f ? a1 : 0.0f;
        b.x = W2[ka * 32 + n0 + r];
        b.y = W2[(ka + 1) * 32 + n0 + r];
        acc = __builtin_amdgcn_wmma_f32_16x16x4_f32(
            /*neg_a=*/false, a, /*neg_b=*/false, b,
            /*c_mod=*/(short)0, acc, /*reuse_a=*/false, /*reuse_b=*/false);
    }

    int col = n0 + r;
    int rowBase = m0 + khalf * 8;
#pragma unroll
    for (int v = 0; v < 8; ++v) {
        h2pre[(rowBase + v) * 32 + col] = acc[v];
    }
}

// Layer-2 edge aggregation: one wave per edge, lane = feature (32 feats = wave32).
// Coalesced 128B gather of h2pre[src,:], 32 L2-resident atomics into agg2[dst,:].
__global__ void gcn_edge2(const int* __restrict__ src, const int* __restrict__ dst,
                          const float* __restrict__ dis,
                          const float* __restrict__ h2pre,
                          float* __restrict__ agg2, int E) {
    int lane = threadIdx.x & 31;
    int wave = (int)((blockIdx.x * blockDim.x + threadIdx.x) >> 5);
    int nWaves = (int)((gridDim.x * blockDim.x) >> 5);
    for (int e = wave; e < E; e += nWaves) {
        int s = src[e], d = dst[e];
        float norm = dis[s] * dis[d];
        float v = h2pre[s * 32 + lane] * norm;
        atomicAdd(&agg2[d * 32 + lane], v);
    }
}

// Fused layer-2 epilogue (self-loop + bias + relu) and global mean-pool sums.
__global__ void gcn_pool(const float* __restrict__ agg2, const float* __restrict__ h2pre,
                         const float* __restrict__ dis, const float* __restrict__ b2,
                         const int* __restrict__ batch,
                         float* __restrict__ psum, float* __restrict__ pcnt, int N) {
    long tid = (long)blockIdx.x * blockDim.x + threadIdx.x;
    long total = (long)N * 32;
    long stride = (long)gridDim.x * blockDim.x;
    for (; tid < total; tid += stride) {
        int i = (int)(tid >> 5);
        int f = (int)(tid & 31);
        float d2 = dis[i] * dis[i];
        float h = agg2[tid] + d2 * h2pre[tid] + b2[f];
        h = h > 0.0f ? h : 0.0f;
        int g = batch[i];
        atomicAdd(&psum[g * 32 + f], h);
        if (f == 0) atomicAdd(&pcnt[g], 1.0f);
    }
}

// pooled = psum / max(pcnt,1); out = relu(pooled@Wp1 + bp1) @ Wp2 + bp2
__global__ void gcn_mlp(const float* __restrict__ psum, const float* __restrict__ pcnt,
                        const float* __restrict__ Wp1, const float* __restrict__ bp1,
                        const float* __restrict__ Wp2, const float* __restrict__ bp2,
                        float* __restrict__ out, int G) {
    int g = blockIdx.x * blockDim.x + threadIdx.x;
    if (g >= G) return;
    float c = pcnt[g];
    c = c > 1.0f ? c : 1.0f;
    float inv = 1.0f / c;
    float pooled[32];
#pragma unroll
    for (int f = 0; f < 32; ++f) pooled[f] = psum[g * 32 + f] * inv;
    float o0 = bp2[0], o1 = bp2[1], o2 = bp2[2];
    for (int j = 0; j < 128; ++j) {
        float h = bp1[j];
#pragma unroll
        for (int f = 0; f < 32; ++f) h = fmaf(pooled[f], Wp1[f * 128 + j], h);
        h = h > 0.0f ? h : 0.0f;
        o0 = fmaf(h, Wp2[j * 3 + 0], o0);
        o1 = fmaf(h, Wp2[j * 3 + 1], o1);
        o2 = fmaf(h, Wp2[j * 3 + 2], o2);
    }
    out[g * 3 + 0] = o0;
    out[g * 3 + 1] = o1;
    out[g * 3 + 2] = o2;
}

extern "C" void kernel_launch(void* const* d_in, const int* in_sizes, int n_in,
                              void* d_out, int out_size, void* d_ws, size_t ws_size,
                              hipStream_t stream) {
    const float* x   = (const float*)d_in[0];
    const float* W1  = (const float*)d_in[1];
    const float* b1  = (const float*)d_in[2];
    const float* W2  = (const float*)d_in[3];
    const float* b2  = (const float*)d_in[4];
    const float* Wp1 = (const float*)d_in[5];
    const float* bp1 = (const float*)d_in[6];
    const float* Wp2 = (const float*)d_in[7];
    const float* bp2 = (const float*)d_in[8];
    const int* eidx  = (const int*)d_in[9];
    const int* batch = (const int*)d_in[10];
    float* out = (float*)d_out;

    const int N = in_sizes[0];
    const int E = in_sizes[9] / 2;
    const int G = out_size / 3;
    const int* src = eidx;
    const int* dst = eidx + E;

    float* wsf   = (float*)d_ws;
    float* deg   = wsf;
    float* s1    = deg + N;
    float* dis   = s1 + N;
    float* tbuf  = dis + N;
    float* h2pre = tbuf + N;
    float* agg2  = h2pre + (long)N * 32;
    float* psum  = agg2 + (long)N * 32;
    float* pcnt  = psum + (long)G * 32;

    const int TPB = 256;

    // zero: [deg, s1] (2N contiguous) and [agg2, psum, pcnt] (contiguous)
    long nz1 = 2L * N;
    long nz2 = 32L * N + 32L * G + G;
    gcn_zero_f32<<<1024, TPB, 0, stream>>>(deg, nz1);
    gcn_zero_f32<<<2048, TPB, 0, stream>>>(agg2, nz2);

    gcn_degree<<<(E + TPB - 1) / TPB, TPB, 0, stream>>>(dst, deg, E);
    gcn_dis<<<(N + TPB - 1) / TPB, TPB, 0, stream>>>(deg, dis, N);
    gcn_edge1<<<(E + TPB - 1) / TPB, TPB, 0, stream>>>(src, dst, x, dis, s1, E);
    gcn_make_t<<<(N + TPB - 1) / TPB, TPB, 0, stream>>>(s1, dis, x, tbuf, N);

    int nTiles = (N / 16) * 2;  // N is 100000 (multiple of 16); 2 col tiles
    gcn_gemm_wmma<<<(nTiles + 7) / 8, TPB, 0, stream>>>(tbuf, W1, b1, W2, h2pre, nTiles);

    gcn_edge2<<<4096, TPB, 0, stream>>>(src, dst, dis, h2pre, agg2, E);

    gcn_pool<<<4096, TPB, 0, stream>>>(agg2, h2pre, dis, b2, batch, psum, pcnt, N);

    gcn_mlp<<<(G + TPB - 1) / TPB, TPB, 0, stream>>>(psum, pcnt, Wp1, bp1, Wp2, bp2, out, G);
}